// Leaf_Integ_16097537425998
// MI455X (gfx1250) — compile-verified
//
#include <hip/hip_runtime.h>

// ---------------------------------------------------------------------------
// Leaf_Integ scan for MI455X (gfx1250).
//
// 1024 independent channels; each runs a 4096-step recurrence whose 64-tap
// per-channel history filter is evaluated EXACTLY via sliding windowed
// moment sums: kern[j-1] = (c1*j + c0)*rho^j for j >= jm (zero below,
// reproducing the reference's max(t-delta,0) clamp), so
//   filtered_t = sum_b c1_b*V_b + c0_b*U_b,
//   U_b = sum_{j=jm..64} Z_{t-j} rho^j,  V_b = sum_{j=jm..64} j Z_{t-j} rho^j,
// slid per step with the j=65 exit term subtracted using the 64-deep ring
// (same finite sum as the reference, re-associated; updates contract).
// The 4 bases are held as two <2 x float> vectors and updated with
// __builtin_elementwise_fma so the recurrence lowers to v_pk_fma_f32.
//
// waves_per_eu(1,1) is the chosen register/scheduling config for
// this kernel (one wave per SIMD exists anyway; default budgets sprawl more).
//
// Depthwise per-channel weights => no GEMM structure => WMMA inapplicable;
// compute is fp32 VALU. CDNA5 async global->LDS prefetch (ASYNCcnt path)
// streams the three inputs 16 steps ahead of the serial scan; one
// ds_load_b128 per step consumes a staged (S, up, noise) triple.
// ---------------------------------------------------------------------------

#define T_DATA   4096
#define SUBN     1024
#define T_HIST   64
#define DEPTH    16         // prefetch distance (steps)
#define SLOTS    32         // LDS ring slots (2x DEPTH; divides 64 -> static idx)
#define GROUP    4          // steps consumed per s_wait_asynccnt

typedef float v2f __attribute__((ext_vector_type(2)));

// CDNA5 async copy: global memory -> LDS, tracked by ASYNCcnt.
// %0 = VGPR with LDS byte address, %1 = VGPR pair with 64-bit global address.
__device__ __forceinline__ void async_ld32(const float* g, unsigned lds_off) {
    asm volatile("global_load_async_to_lds_b32 %0, %1, off"
                 :: "v"(lds_off), "v"(g));
}

// Low 32 bits of a generic pointer into __shared__ == wave-relative LDS addr.
__device__ __forceinline__ unsigned lds_off_of(const void* p) {
    return (unsigned)(unsigned long long)p;
}

__device__ __forceinline__ float sigf(float x) {
    return __builtin_amdgcn_rcpf(1.0f + __expf(-x));
}

__global__ __launch_bounds__(32)
__attribute__((amdgpu_waves_per_eu(1, 1)))
void leaf_integ_scan(const float* __restrict__ S_conv,
                     const float* __restrict__ up_mu,
                     const float* __restrict__ noise,
                     const float* __restrict__ W_sub,
                     const float* __restrict__ th_syn,
                     const float* __restrict__ th_spk,
                     const float* __restrict__ W_spk,
                     const float* __restrict__ tau_hist,
                     const float* __restrict__ K_hist,
                     const float* __restrict__ delta_hist,
                     float* __restrict__ out)
{
    // [slot][lane] = {S, up, noise, pad}; async B32 x3 in, ds_load_b128 out.
    __shared__ float4 stage[SLOTS][32];     // 16 KB

    const int lane = threadIdx.x;           // 0..31 (wave32)
    const int s    = blockIdx.x * 32 + lane;

    // ---- async prefetch prologue: steps 0..DEPTH-1 ----------------------
    #pragma unroll
    for (int t = 0; t < DEPTH; ++t) {
        const size_t o = (size_t)t * SUBN + s;
        const unsigned base = lds_off_of(&stage[t][lane]);
        async_ld32(S_conv + o, base + 0);
        async_ld32(up_mu  + o, base + 4);
        async_ld32(noise  + o, base + 8);
    }

    // ---- per-channel constants ------------------------------------------
    const float wsub  = W_sub[s];
    const float tsyn  = th_syn[s];
    const float tspk  = th_spk[s];
    const float wspk  = W_spk[s];
    const float delta = delta_hist[s];

    // Window-entry lag: kern[j-1] = 0 for j < jm = ceil(delta+1); delta in [0,3).
    const float jmf = ceilf(delta + 1.0f);            // 1..4
    const int   jm  = (int)jmf;
    const bool  j1  = (jm == 1), j2 = (jm == 2), j3 = (jm == 3);

    // Basis coefficients, packed 2 bases per <2 x float>.
    v2f rho01, rho23, piu01, piu23, piv01, piv23;
    v2f pou01, pou23, pov01, pov23, c1_01, c1_23, c0_01, c0_23;
    #pragma unroll
    for (int b = 0; b < 4; ++b) {
        const float inv = __expf(-tau_hist[b]);       // 1/tau_b, tau_b=exp(tau_hist[b])
        const float kb  = K_hist[s * 4 + b];
        const float rh  = __expf(-inv);
        const float cc1 = kb * inv * __expf((1.0f + delta) * inv);
        const float cc0 = -(1.0f + delta) * cc1;
        const float pou = __expf(-65.0f * inv);       // rho^65 (exit, j=65)
        const float pov = 65.0f * pou;
        const float piu = __expf(-jmf * inv);         // rho^jm (entry, j=jm)
        const float piv = jmf * piu;
        const int e = b & 1;
        if (b < 2) { rho01[e]=rh; c1_01[e]=cc1; c0_01[e]=cc0;
                     pou01[e]=pou; pov01[e]=pov; piu01[e]=piu; piv01[e]=piv; }
        else       { rho23[e]=rh; c1_23[e]=cc1; c0_23[e]=cc0;
                     pou23[e]=pou; pov23[e]=pov; piu23[e]=piu; piv23[e]=piv; }
    }

    // Windowed moment state (zero history => zero sums).
    v2f U01 = {0.f, 0.f}, U23 = {0.f, 0.f};
    v2f V01 = {0.f, 0.f}, V23 = {0.f, 0.f};

    // history ring: z[m] holds Z_{t'} with t' == m (mod 64); starts at zero.
    float z[T_HIST];
    #pragma unroll
    for (int i = 0; i < T_HIST; ++i) z[i] = 0.0f;

    const size_t N = (size_t)T_DATA * SUBN;
    float* __restrict__ outY = out;
    float* __restrict__ outZ = out + N;
    float* __restrict__ outM = out + 2 * N;
    float* __restrict__ outD = out + 3 * N;

    for (int T0 = 0; T0 < T_DATA; T0 += T_HIST) {
        #pragma unroll
        for (int pg = 0; pg < T_HIST; pg += GROUP) {
            // Steps [T0+pg, T0+pg+3] are done when ASYNCcnt <= 3*(DEPTH-GROUP)
            // = 36 (async loads complete in order; 3 issues per step).
            asm volatile("s_wait_asynccnt 36" ::: "memory");

            // Consume: one ds_load_b128 per step; slot = p & 31 (static,
            // since T0 is a multiple of 64 and SLOTS divides 64).
            float4 in[GROUP];
            #pragma unroll
            for (int q = 0; q < GROUP; ++q)
                in[q] = stage[(pg + q) & (SLOTS - 1)][lane];

            // Re-issue DEPTH steps ahead; power-of-two wrap keeps the tail
            // branch-free (dummy loads land in slots never read again).
            #pragma unroll
            for (int q = 0; q < GROUP; ++q) {
                const int ti  = (T0 + pg + q + DEPTH) & (T_DATA - 1);
                const int isl = (pg + q + DEPTH) & (SLOTS - 1);
                const size_t o = (size_t)ti * SUBN + s;
                const unsigned base = lds_off_of(&stage[isl][lane]);
                async_ld32(S_conv + o, base + 0);
                async_ld32(up_mu  + o, base + 4);
                async_ld32(noise  + o, base + 8);
            }

            // Compute the 4 steps (serial recurrence).
            #pragma unroll
            for (int q = 0; q < GROUP; ++q) {
                const int p = pg + q;               // t mod 64 (static)
                const int t = T0 + p;
                const float s_t  = in[q].x;
                const float up_t = in[q].y;
                const float n_t  = in[q].z;

                // filtered_t = sum_b c1[b]*V[b] + c0[b]*U[b]   (v_pk_fma_f32)
                const v2f fa = __builtin_elementwise_fma(c1_01, V01, c0_01 * U01);
                const v2f fb = __builtin_elementwise_fma(c1_23, V23, c0_23 * U23);
                const v2f fs = fa + fb;
                const float filtered = fs.x + fs.y;

                const float X_in  = s_t + filtered + tsyn;
                const float X_out = sigf(X_in);
                const float Y     = X_out * wsub;
                const float down  = fmaf(X_out, wspk, tspk);
                const float mu    = 0.5f * (up_t + down);
                const float Z     = sigf(mu + n_t);   // sigma = 1

                // Slide the window t -> t+1 (exact: subtract the j=65 exit
                // term, add the j=jm entry term Z_{t+1-jm}).
                const float zout = z[p];              // Z_{t-64}
                const float zin  = j1 ? Z
                                 : (j2 ? z[(p - 1) & 63]
                                 : (j3 ? z[(p - 2) & 63]
                                 :       z[(p - 3) & 63]));
                const v2f zi = {zin, zin};
                const v2f zo = {zout, zout};
                const v2f wu01 = __builtin_elementwise_fma(piu01, zi, -(pou01 * zo));
                const v2f wu23 = __builtin_elementwise_fma(piu23, zi, -(pou23 * zo));
                const v2f wv01 = __builtin_elementwise_fma(piv01, zi, -(pov01 * zo));
                const v2f wv23 = __builtin_elementwise_fma(piv23, zi, -(pov23 * zo));
                V01 = __builtin_elementwise_fma(rho01, V01 + U01, wv01);
                V23 = __builtin_elementwise_fma(rho23, V23 + U23, wv23);
                U01 = __builtin_elementwise_fma(rho01, U01, wu01);
                U23 = __builtin_elementwise_fma(rho23, U23, wu23);

                z[p] = Z;  // overwrite lag-64 slot (t mod 64 == p)

                const size_t o = (size_t)t * SUBN + s;
                outY[o] = Y;
                outZ[o] = Z;
                outM[o] = mu;
                outD[o] = down;
            }
        }
    }
}

extern "C" void kernel_launch(void* const* d_in, const int* in_sizes, int n_in,
                              void* d_out, int out_size, void* d_ws, size_t ws_size,
                              hipStream_t stream) {
    (void)in_sizes; (void)n_in; (void)out_size; (void)d_ws; (void)ws_size;
    leaf_integ_scan<<<dim3(SUBN / 32), dim3(32), 0, stream>>>(
        (const float*)d_in[0],  // S_conv   (4096,1024)
        (const float*)d_in[1],  // up_mu_Z  (4096,1024)
        (const float*)d_in[2],  // noise    (4096,1024)
        (const float*)d_in[3],  // W_sub    (1024)
        (const float*)d_in[4],  // theta_syn(1024)
        (const float*)d_in[5],  // theta_spike(1024)
        (const float*)d_in[6],  // W_spike  (1024)
        (const float*)d_in[7],  // tau_hist (4)
        (const float*)d_in[8],  // K_hist   (1024,4)
        (const float*)d_in[9],  // delta_hist(1024)
        (float*)d_out);         // [Y | Z | mu_Z | down_mu_Z], each 4096*1024 f32
}